// LIFNeuron_62285615727463
// MI455X (gfx1250) — compile-verified
//
#include <hip/hip_runtime.h>
#include <stdint.h>

// LIF scan over T for [B,F,T] fp32, memory-bound (floor ~22us @ 23.3 TB/s).
// Strategy: async-stage [128 neurons x 32 t] tiles into LDS with
// global_load_async_to_lds_b128 (ASYNCcnt), double-buffered; per-thread scan
// from LDS; coalesced b128 drain of spikes back to global.

#define T_TOTAL    512
#define TCHUNK     32
#define NCHUNK     (T_TOTAL / TCHUNK)            // 16
#define NTHREADS   128                           // 4 wave32
#define ROWS       NTHREADS                      // neurons per block
#define ROWF       (TCHUNK + 4)                  // 36 floats padded stride
#define ROWB       (ROWF * 4)                    // 144 bytes (multiple of 16)
#define BUF_FLOATS (ROWS * ROWF)                 // 4608 floats = 18432 B
#define BUF_BYTES  (BUF_FLOATS * 4)
#define PIECES_PER_ROW     (TCHUNK * 4 / 16)     // 8 x 16B pieces per row
#define PIECES_PER_THREAD  (ROWS * PIECES_PER_ROW / NTHREADS)  // 8

#define INV_TAU 0.05f
#define THRESH  1.0f

__global__ __launch_bounds__(NTHREADS)
void lif_scan_kernel(const float* __restrict__ x, float* __restrict__ out)
{
    __shared__ __align__(16) float smem[2 * BUF_FLOATS];   // 36 KB, double buffer

    const int tid = threadIdx.x;
    const uint32_t n0 = (uint32_t)blockIdx.x * ROWS;       // first neuron of block
    const uint64_t xbase = (uint64_t)(uintptr_t)x;
    // Low 32 bits of a generic pointer to LDS are the LDS byte offset.
    const uint32_t lds0 = (uint32_t)(uintptr_t)&smem[0];

    // Issue one tile's worth of async global->LDS copies.
    // Piece q = j*NTHREADS + tid; row r = q/8; 16B piece p = q%8.
    // 8 consecutive lanes cover one neuron's 128B time-chunk -> full-line reads.
    auto issue_loads = [&](int tc, int b) {
        const uint32_t lbase = lds0 + (uint32_t)b * BUF_BYTES;
        const uint32_t gchunk = (uint32_t)tc * (TCHUNK * 4);   // byte offset in row
#pragma unroll
        for (int j = 0; j < PIECES_PER_THREAD; ++j) {
            const int q = j * NTHREADS + tid;
            const int r = q >> 3;
            const int p = q & 7;
            const uint32_t goff = (n0 + (uint32_t)r) * (T_TOTAL * 4) + gchunk + (uint32_t)p * 16;
            const uint32_t loff = lbase + (uint32_t)r * ROWB + (uint32_t)p * 16;
            asm volatile("global_load_async_to_lds_b128 %0, %1, %2"
                         :: "v"(loff), "v"(goff), "s"(xbase)
                         : "memory");
        }
    };

    float v = 0.0f;   // membrane potential, carried across all chunks

    issue_loads(0, 0);

    for (int c = 0; c < NCHUNK; ++c) {
        const int b = c & 1;

        // Wait for this chunk's async loads (this wave), then make visible.
        asm volatile("s_wait_asynccnt 0x0" ::: "memory");
        __syncthreads();

        // Prefetch next chunk into the other buffer (overlaps compute below).
        if (c + 1 < NCHUNK) issue_loads(c + 1, b ^ 1);

        // Sequential LIF scan of this thread's neuron row, in-place spike write.
        float* row = &smem[b * BUF_FLOATS + tid * ROWF];
#pragma unroll
        for (int g = 0; g < TCHUNK / 4; ++g) {
            float4 xv = *(float4*)(row + 4 * g);
            float4 s;
#define LIF_STEP(XX, SS)                                   \
            do {                                           \
                v += (XX - v) * INV_TAU;                   \
                const bool fire = (v >= THRESH);           \
                SS = fire ? 1.0f : 0.0f;                   \
                v  = fire ? 0.0f : v;                      \
            } while (0)
            LIF_STEP(xv.x, s.x);
            LIF_STEP(xv.y, s.y);
            LIF_STEP(xv.z, s.z);
            LIF_STEP(xv.w, s.w);
#undef LIF_STEP
            *(float4*)(row + 4 * g) = s;   // overwrite consumed input with spikes
        }

        __syncthreads();   // all spike writes visible before coalesced drain

        // Coalesced drain: same piece mapping as loads -> full-line stores.
        const int lbase = b * BUF_FLOATS;
#pragma unroll
        for (int j = 0; j < PIECES_PER_THREAD; ++j) {
            const int q = j * NTHREADS + tid;
            const int r = q >> 3;
            const int p = q & 7;
            const float4 sv = *(const float4*)&smem[lbase + r * ROWF + p * 4];
            *(float4*)(out + (size_t)(n0 + (uint32_t)r) * T_TOTAL
                           + (size_t)c * TCHUNK + (size_t)p * 4) = sv;
        }
        // Next iteration's leading barrier (after s_wait_asynccnt) separates this
        // drain from chunk c+2's async loads into the same buffer.
    }
}

extern "C" void kernel_launch(void* const* d_in, const int* in_sizes, int n_in,
                              void* d_out, int out_size, void* d_ws, size_t ws_size,
                              hipStream_t stream) {
    const float* x = (const float*)d_in[0];
    float* out = (float*)d_out;

    const int total    = in_sizes[0];          // B*F*T
    const int nNeurons = total / T_TOTAL;      // B*F = 131072
    const int blocks   = nNeurons / ROWS;      // 1024

    if (blocks > 0) {
        lif_scan_kernel<<<blocks, NTHREADS, 0, stream>>>(x, out);
    }
}